// TextDecoder_17755394802151
// MI455X (gfx1250) — compile-verified
//
#include <hip/hip_runtime.h>
#include <hip/hip_bf16.h>
#include <math.h>

#define VOCAB 32000
#define EMB   512
#define HID   1024
#define G3H   3072
#define BATCH 32
#define TLEN  128

typedef float v2f __attribute__((ext_vector_type(2)));
typedef float v8f __attribute__((ext_vector_type(8)));

// D = A(16x4 f32) * B(4x16 f32) + C(16x16 f32), exact fp32 MAC path on CDNA5.
__device__ __forceinline__ v8f wmma_f32_16x16x4(v2f a, v2f b, v8f c) {
  return __builtin_amdgcn_wmma_f32_16x16x4_f32(false, a, false, b, (short)0, c,
                                               false, false);
}

// ---------------------------------------------------------------------------
// init: h = encoder_hidden, tok = 1
// ---------------------------------------------------------------------------
__global__ __launch_bounds__(256) void k_init(const float* __restrict__ enc,
                                              float* __restrict__ h,
                                              int* __restrict__ tok) {
  int i = blockIdx.x * 256 + threadIdx.x;   // exactly BATCH*HID threads
  h[i] = enc[i];
  if (i < BATCH) tok[i] = 1;
}

// ---------------------------------------------------------------------------
// gates: gi = embedding[tok] @ W_ih^T + b_ih   (blockIdx.y == 0, K = 512)
//        gh = h              @ W_hh^T + b_hh   (blockIdx.y == 1, K = 1024)
// Block = 128 threads = 4 waves; each wave computes a 32(M) x 32(N) tile
// (2 M-tiles x 2 N-tiles -> 4 WMMA per k-step on 2 A + 2 B loads, 1:1 ratio).
// ---------------------------------------------------------------------------
__global__ __launch_bounds__(128) void k_gates(const float* __restrict__ emb,
                                               const int* __restrict__ tok,
                                               const float* __restrict__ h,
                                               const float* __restrict__ W_ih,
                                               const float* __restrict__ W_hh,
                                               const float* __restrict__ b_ih,
                                               const float* __restrict__ b_hh,
                                               float* __restrict__ gi,
                                               float* __restrict__ gh) {
  const int lane = threadIdx.x & 31;
  const int half = lane >> 4;         // 0: K=k..k+1, 1: K=k+2..k+3
  const int l    = lane & 15;         // M row (A) / N col (B,C)
  const int wave = threadIdx.x >> 5;
  const int n0   = blockIdx.x * 128 + wave * 32;

  const float *a0, *a1, *W, *bias;
  float* out;
  int K;
  if (blockIdx.y == 0) {              // input gates: gather rows of embedding
    a0 = emb + (size_t)tok[l] * EMB;
    a1 = emb + (size_t)tok[16 + l] * EMB;
    W = W_ih; bias = b_ih; out = gi; K = EMB;
  } else {                            // hidden gates
    a0 = h + (size_t)l * HID;
    a1 = h + (size_t)(16 + l) * HID;
    W = W_hh; bias = b_hh; out = gh; K = HID;
  }
  const float* brow0 = W + (size_t)(n0 + l) * K;
  const float* brow1 = W + (size_t)(n0 + 16 + l) * K;
  const int ko = half * 2;

  v8f c00 = {0.f,0.f,0.f,0.f,0.f,0.f,0.f,0.f};
  v8f c01 = c00, c10 = c00, c11 = c00;

  #pragma unroll 4
  for (int k = 0; k < K; k += 4) {
    v2f af0 = *(const v2f*)(a0 + k + ko);      // A rows 0..15
    v2f af1 = *(const v2f*)(a1 + k + ko);      // A rows 16..31
    v2f bf0 = *(const v2f*)(brow0 + k + ko);   // B cols n0..n0+15
    v2f bf1 = *(const v2f*)(brow1 + k + ko);   // B cols n0+16..n0+31
    c00 = wmma_f32_16x16x4(af0, bf0, c00);
    c10 = wmma_f32_16x16x4(af1, bf0, c10);
    c01 = wmma_f32_16x16x4(af0, bf1, c01);
    c11 = wmma_f32_16x16x4(af1, bf1, c11);
  }

  const float bj0 = bias[n0 + l];
  const float bj1 = bias[n0 + 16 + l];
  #pragma unroll
  for (int r = 0; r < 8; ++r) {
    int m = r + half * 8;               // C layout: VGPR r -> M = r (+8 hi half)
    out[(size_t)m * G3H + n0 + l]             = c00[r] + bj0;
    out[(size_t)(16 + m) * G3H + n0 + l]      = c10[r] + bj0;
    out[(size_t)m * G3H + n0 + 16 + l]        = c01[r] + bj1;
    out[(size_t)(16 + m) * G3H + n0 + 16 + l] = c11[r] + bj1;
  }
}

// ---------------------------------------------------------------------------
// fuse: PyTorch GRU gate math, h updated in place
// ---------------------------------------------------------------------------
__global__ __launch_bounds__(256) void k_fuse(const float* __restrict__ gi,
                                              const float* __restrict__ gh,
                                              float* __restrict__ h) {
  int i = blockIdx.x * 256 + threadIdx.x;   // exactly BATCH*HID threads
  int b = i >> 10;
  int j = i & (HID - 1);
  const float* gib = gi + (size_t)b * G3H;
  const float* ghb = gh + (size_t)b * G3H;
  float r = 1.f / (1.f + __expf(-(gib[j] + ghb[j])));
  float z = 1.f / (1.f + __expf(-(gib[HID + j] + ghb[HID + j])));
  float n = tanhf(gib[2 * HID + j] + r * ghb[2 * HID + j]);
  h[i] = (1.f - z) * n + z * h[i];
}

// ---------------------------------------------------------------------------
// logits: pred[b, t, v] = h @ W_out^T + b_out ; W_out is L2-resident (131 MB)
// Block = 128 threads = 4 waves; each wave computes a 32(M) x 32(N) tile.
// ---------------------------------------------------------------------------
__global__ __launch_bounds__(128) void k_logits(const float* __restrict__ h,
                                                const float* __restrict__ W_out,
                                                const float* __restrict__ b_out,
                                                float* __restrict__ out,
                                                int t) {
  const int lane = threadIdx.x & 31;
  const int half = lane >> 4;
  const int l    = lane & 15;
  const int wave = threadIdx.x >> 5;
  const int n0   = blockIdx.x * 128 + wave * 32;

  const float* a0    = h + (size_t)l * HID;
  const float* a1    = h + (size_t)(16 + l) * HID;
  const float* brow0 = W_out + (size_t)(n0 + l) * HID;
  const float* brow1 = W_out + (size_t)(n0 + 16 + l) * HID;
  const int ko = half * 2;

  v8f c00 = {0.f,0.f,0.f,0.f,0.f,0.f,0.f,0.f};
  v8f c01 = c00, c10 = c00, c11 = c00;

  for (int kc = 0; kc < HID; kc += 64) {
    // pull next weight chunk toward the WGP (global_prefetch_b8)
    if (kc + 64 < HID) {
      __builtin_prefetch(brow0 + kc + 64, 0, 3);
      __builtin_prefetch(brow1 + kc + 64, 0, 3);
    }
    #pragma unroll
    for (int k = kc; k < kc + 64; k += 4) {
      v2f af0 = *(const v2f*)(a0 + k + ko);
      v2f af1 = *(const v2f*)(a1 + k + ko);
      v2f bf0 = *(const v2f*)(brow0 + k + ko);
      v2f bf1 = *(const v2f*)(brow1 + k + ko);
      c00 = wmma_f32_16x16x4(af0, bf0, c00);
      c10 = wmma_f32_16x16x4(af1, bf0, c10);
      c01 = wmma_f32_16x16x4(af0, bf1, c01);
      c11 = wmma_f32_16x16x4(af1, bf1, c11);
    }
  }

  const float bj0 = b_out[n0 + l];
  const float bj1 = b_out[n0 + 16 + l];
  #pragma unroll
  for (int r = 0; r < 8; ++r) {
    int m = r + half * 8;
    out[(((size_t)m        * TLEN) + t) * VOCAB + n0 + l]      = c00[r] + bj0;
    out[(((size_t)(16 + m) * TLEN) + t) * VOCAB + n0 + l]      = c10[r] + bj0;
    out[(((size_t)m        * TLEN) + t) * VOCAB + n0 + 16 + l] = c01[r] + bj1;
    out[(((size_t)(16 + m) * TLEN) + t) * VOCAB + n0 + 16 + l] = c11[r] + bj1;
  }
}

// ---------------------------------------------------------------------------
// argmax over vocab per batch row; first-max tie-break matches jnp.argmax
// ---------------------------------------------------------------------------
__global__ __launch_bounds__(256) void k_argmax(const float* __restrict__ out,
                                                int* __restrict__ tok, int t) {
  __shared__ float sv[256];
  __shared__ int   si[256];
  const int b = blockIdx.x;
  const float* row = out + ((size_t)b * TLEN + t) * VOCAB;

  float best = -INFINITY;
  int   bi   = 0;
  for (int v = threadIdx.x; v < VOCAB; v += 256) {
    float x = row[v];
    if (x > best) { best = x; bi = v; }   // per-thread indices increase -> first max
  }
  sv[threadIdx.x] = best;
  si[threadIdx.x] = bi;
  __syncthreads();
  for (int s = 128; s > 0; s >>= 1) {
    if (threadIdx.x < s) {
      float ov = sv[threadIdx.x + s];
      int   oi = si[threadIdx.x + s];
      if (ov > sv[threadIdx.x] ||
          (ov == sv[threadIdx.x] && oi < si[threadIdx.x])) {
        sv[threadIdx.x] = ov;
        si[threadIdx.x] = oi;
      }
    }
    __syncthreads();
  }
  if (threadIdx.x == 0) tok[b] = si[0];
}

// ---------------------------------------------------------------------------
extern "C" void kernel_launch(void* const* d_in, const int* in_sizes, int n_in,
                              void* d_out, int out_size, void* d_ws,
                              size_t ws_size, hipStream_t stream) {
  const float* enc   = (const float*)d_in[0];
  const float* emb   = (const float*)d_in[1];
  const float* W_ih  = (const float*)d_in[2];
  const float* W_hh  = (const float*)d_in[3];
  const float* b_ih  = (const float*)d_in[4];
  const float* b_hh  = (const float*)d_in[5];
  const float* W_out = (const float*)d_in[6];
  const float* b_out = (const float*)d_in[7];
  float* out = (float*)d_out;

  // workspace layout (~0.92 MB): h | gi | gh | tok
  float* h  = (float*)d_ws;
  float* gi = h + BATCH * HID;
  float* gh = gi + BATCH * G3H;
  int*   tok = (int*)(gh + BATCH * G3H);

  k_init<<<(BATCH * HID) / 256, 256, 0, stream>>>(enc, h, tok);
  for (int t = 0; t < TLEN; ++t) {
    k_gates<<<dim3(G3H / 128, 2), 128, 0, stream>>>(emb, tok, h, W_ih, W_hh,
                                                    b_ih, b_hh, gi, gh);
    k_fuse<<<(BATCH * HID) / 256, 256, 0, stream>>>(gi, gh, h);
    k_logits<<<VOCAB / 128, 128, 0, stream>>>(h, W_out, b_out, out, t);
    k_argmax<<<BATCH, 256, 0, stream>>>(out, tok, t);
  }
}